// M_Metric_49271864820312
// MI455X (gfx1250) — compile-verified
//
#include <hip/hip_runtime.h>

#define DIM   256
#define BATCH 16
#define NPTS  256

typedef float v2f __attribute__((ext_vector_type(2)));
typedef float v8f __attribute__((ext_vector_type(8)));

// ---------------------------------------------------------------------------
// Kernel 1: Y[b] = X (N x DIM) @ M[b] (DIM x DIM), fp32 WMMA 16x16x4.
// One wave per 16x32 tile of Y[b]: two independent accumulators share the
// A fragment -> 2 interleaved WMMA chains, half the A traffic.
// EXEC all-1s (32 active lanes) as WMMA requires.
// ---------------------------------------------------------------------------
__global__ __launch_bounds__(32) void k_gemm_xm(const float* __restrict__ X,
                                                const float* __restrict__ M,
                                                float* __restrict__ Y) {
  const int i0 = blockIdx.x * 16;          // output row tile
  const int j0 = blockIdx.y * 32;          // output col tile (2 x 16)
  const int b  = blockIdx.z;               // batch
  const float* __restrict__ Mb = M + (size_t)b * DIM * DIM;
  float* __restrict__ Yb = Y + (size_t)b * NPTS * DIM;

  const int lane = threadIdx.x;            // 0..31
  const int half = lane >> 4;              // 0 or 1
  const int lm   = lane & 15;

  v8f acc0 = {};
  v8f acc1 = {};
  for (int k = 0; k < DIM; k += 4) {
    // A fragment: A[m][kk] -> lane = m + 16*(kk/2), vgpr = kk&1 (contig float2)
    v2f a = *(const v2f*)&X[(size_t)(i0 + lm) * DIM + k + 2 * half];
    // B fragments: B[kk][n] -> lane = n + 16*(kk/2), vgpr = kk&1
    const size_t r0 = (size_t)(k + 2 * half + 0) * DIM + j0 + lm;
    const size_t r1 = (size_t)(k + 2 * half + 1) * DIM + j0 + lm;
    v2f b0, b1;
    b0.x = Mb[r0];      b0.y = Mb[r1];        // cols j0..j0+15
    b1.x = Mb[r0 + 16]; b1.y = Mb[r1 + 16];   // cols j0+16..j0+31
    acc0 = __builtin_amdgcn_wmma_f32_16x16x4_f32(
        false, a, false, b0, (short)0, acc0, false, false);
    acc1 = __builtin_amdgcn_wmma_f32_16x16x4_f32(
        false, a, false, b1, (short)0, acc1, false, false);
  }

#pragma unroll
  for (int v = 0; v < 8; ++v) {
    const int row = i0 + v + 8 * half;     // D layout: vgpr v, lane half
    Yb[(size_t)row * DIM + j0 + lm]      = acc0[v];
    Yb[(size_t)row * DIM + j0 + 16 + lm] = acc1[v];
  }
}

// ---------------------------------------------------------------------------
// Kernel 2: q[b][i] = dot(X[i,:], Y[b,i,:])  (the diagonal of X M_b X^T).
// One wave per (b,i); coalesced along d; wave32 shuffle reduction.
// ---------------------------------------------------------------------------
__global__ __launch_bounds__(256) void k_diag_q(const float* __restrict__ X,
                                                const float* __restrict__ Y,
                                                float* __restrict__ q) {
  const int lane   = threadIdx.x & 31;
  const int waveIn = threadIdx.x >> 5;                 // 0..7
  const int gid    = blockIdx.x * 8 + waveIn;          // wave id = (b, i)
  const int b = gid / NPTS;
  const int i = gid % NPTS;

  const float* __restrict__ xr = X + (size_t)i * DIM;
  const float* __restrict__ yr = Y + ((size_t)b * NPTS + i) * DIM;

  float s = 0.f;
  for (int d = lane; d < DIM; d += 32) s += xr[d] * yr[d];

#pragma unroll
  for (int off = 16; off > 0; off >>= 1) s += __shfl_xor(s, off, 32);

  if (lane == 0) q[(size_t)b * NPTS + i] = s;
}

// ---------------------------------------------------------------------------
// Kernel 3: G_b = Y_b @ X^T via WMMA (16x32 tile per wave), fused epilogue
//           P[b,i,j] = q[b,i] + q[b,j] - 2*G_b[i,j]  (NT streamed out).
// B[k][n] = X[n][k] -> per-lane contiguous float2 load from X row n.
// ---------------------------------------------------------------------------
__global__ __launch_bounds__(32) void k_gemm_yxt(const float* __restrict__ X,
                                                 const float* __restrict__ Y,
                                                 const float* __restrict__ q,
                                                 float* __restrict__ P) {
  const int i0 = blockIdx.x * 16;
  const int j0 = blockIdx.y * 32;
  const int b  = blockIdx.z;
  const float* __restrict__ Yb = Y + (size_t)b * NPTS * DIM;
  const float* __restrict__ qb = q + (size_t)b * NPTS;
  float* __restrict__ Pb = P + (size_t)b * NPTS * NPTS;

  const int lane = threadIdx.x;
  const int half = lane >> 4;
  const int lm   = lane & 15;

  v8f acc0 = {};
  v8f acc1 = {};
  for (int k = 0; k < DIM; k += 4) {
    v2f a  = *(const v2f*)&Yb[(size_t)(i0 + lm)      * DIM + k + 2 * half];
    v2f b0 = *(const v2f*)&X [(size_t)(j0 + lm)      * DIM + k + 2 * half];
    v2f b1 = *(const v2f*)&X [(size_t)(j0 + 16 + lm) * DIM + k + 2 * half];
    acc0 = __builtin_amdgcn_wmma_f32_16x16x4_f32(
        false, a, false, b0, (short)0, acc0, false, false);
    acc1 = __builtin_amdgcn_wmma_f32_16x16x4_f32(
        false, a, false, b1, (short)0, acc1, false, false);
  }

  const int   col0 = j0 + lm;
  const int   col1 = j0 + 16 + lm;
  const float qj0  = qb[col0];
  const float qj1  = qb[col1];
#pragma unroll
  for (int v = 0; v < 8; ++v) {
    const int   row = i0 + v + 8 * half;
    const float qi  = qb[row];
    // P is write-once, never re-read: non-temporal store hint.
    __builtin_nontemporal_store(qi + qj0 - 2.0f * acc0[v],
                                &Pb[(size_t)row * NPTS + col0]);
    __builtin_nontemporal_store(qi + qj1 - 2.0f * acc1[v],
                                &Pb[(size_t)row * NPTS + col1]);
  }
}

// ---------------------------------------------------------------------------
extern "C" void kernel_launch(void* const* d_in, const int* in_sizes, int n_in,
                              void* d_out, int out_size, void* d_ws, size_t ws_size,
                              hipStream_t stream) {
  const float* X = (const float*)d_in[0];   // (N, DIM) fp32
  const float* M = (const float*)d_in[1];   // (B, DIM, DIM) fp32
  float* P = (float*)d_out;                 // (B, N, N) fp32

  float* Y = (float*)d_ws;                                                       // 4 MB
  float* q = (float*)((char*)d_ws + (size_t)BATCH * NPTS * DIM * sizeof(float)); // 16 KB

  dim3 tiles(NPTS / 16, DIM / 32, BATCH);   // (16,8,16) one-wave workgroups
  k_gemm_xm <<<tiles, 32, 0, stream>>>(X, M, Y);
  k_diag_q  <<<(BATCH * NPTS) / 8, 256, 0, stream>>>(X, Y, q);
  k_gemm_yxt<<<tiles, 32, 0, stream>>>(X, Y, q, P);
}